// ST_BIFNeuron_MS_CUDA_81286551044605
// MI455X (gfx1250) — compile-verified
//
#include <hip/hip_runtime.h>

// ST-BIF multi-step neuron, elementwise streaming kernel for gfx1250.
// Pipeline: global_load_async_to_lds_b128 (ASYNCcnt) double-buffered through
// LDS, reduced branch-free 8-step scan in VALU, non-temporal B128 stores.

typedef float v4f __attribute__((ext_vector_type(4)));

#define BLOCK      256          // 8 waves (wave32)
#define TPB        8            // tiles per block
#define TILE_ELEMS (BLOCK * 4)  // floats per tile

// Issue one async DMA: 16 bytes per lane, global -> LDS (GV addressing mode).
// ldsOff is the byte offset within this workgroup's LDS allocation (dynamic
// LDS is the only allocation, so it starts at offset 0).
__device__ __forceinline__ void issue_async_tile(const v4f* __restrict__ xg,
                                                 long tileIdx, unsigned buf,
                                                 unsigned tid) {
    const v4f* gp = xg + tileIdx * BLOCK + tid;
    unsigned ldsOff = (buf * BLOCK + tid) * (unsigned)sizeof(v4f);
    asm volatile("global_load_async_to_lds_b128 %0, %1, off"
                 :
                 : "v"(ldsOff), "v"(gp)
                 : "memory");
}

// 8-step bidirectional integrate-and-fire scan for one element.
// Exact simplifications vs the reference (bit-identical fp32 results):
//  * With T=8 steps and LEVEL=8, |cnt| <= t <= 7 < 8 before every step, so
//    the (cnt < T_max) / (cnt > T_min) clamps can never bind -> removed.
//  * spike*vth is exactly one of {+vth, 0, -vth}; accumulate it directly
//    (acc) instead of carrying the integer spike counter.
// Returns acc / 8 (the time-mean of spike*vth).
__device__ __forceinline__ float stbif_mean(float x, float vth) {
    float v = 0.0f;
    float acc = 0.0f;
#pragma unroll
    for (int t = 0; t < 8; ++t) {
        float h = v + x;                                    // integrate
        float sv = (h >= vth) ? vth : ((h < 0.0f) ? -vth : 0.0f);  // spike*vth
        v = h - sv;                                         // soft reset
        acc += sv;
    }
    return acc * 0.125f;  // exact: divide by T=8
}

__global__ void __launch_bounds__(BLOCK)
stbif_ms_kernel(const float* __restrict__ x,
                const float* __restrict__ qthr,
                float* __restrict__ out,
                int n, int numTiles) {
    extern __shared__ v4f smem[];  // [2][BLOCK] double buffer, 8 KB
    const unsigned tid = threadIdx.x;
    const float vth = qthr[0];

    const v4f* xg = (const v4f*)x;
    v4f* og = (v4f*)out;

    long tile0 = (long)blockIdx.x * TPB;
    int numT = numTiles - (int)tile0;
    numT = (numT > TPB) ? TPB : numT;
    if (numT < 0) numT = 0;

    // Prime the 2-deep pipeline.
    if (numT > 0) issue_async_tile(xg, tile0 + 0, 0u, tid);
    if (numT > 1) issue_async_tile(xg, tile0 + 1, 1u, tid);

    for (int i = 0; i < numT; ++i) {
        // Tile i is done when the outstanding async count drops to the number
        // of tiles still in flight after it. Async loads complete in order
        // per wave, and each lane reads back only the 16 B it DMA'd itself,
        // so a per-wave counter wait suffices -- no barriers.
        if (i + 1 < numT) {
            asm volatile("s_wait_asynccnt 0x1" ::: "memory");
        } else {
            asm volatile("s_wait_asynccnt 0x0" ::: "memory");
        }

        v4f d = smem[(unsigned)(i & 1) * BLOCK + tid];

        v4f r;
        r.x = stbif_mean(d.x, vth);
        r.y = stbif_mean(d.y, vth);
        r.z = stbif_mean(d.z, vth);
        r.w = stbif_mean(d.w, vth);

        // Write-once output: non-temporal B128 store.
        __builtin_nontemporal_store(r, og + (tile0 + i) * BLOCK + tid);

        // Refill the buffer we just drained (the ds_load above was already
        // consumed by the VALU scan, so the WAR on LDS is resolved).
        if (i + 2 < numT) issue_async_tile(xg, tile0 + (long)i + 2, (unsigned)(i & 1), tid);
    }

    // Tail (n % TILE_ELEMS elements), plain scalar path on block 0.
    if (blockIdx.x == 0) {
        for (long idx = (long)numTiles * TILE_ELEMS + tid; idx < (long)n; idx += BLOCK) {
            out[idx] = stbif_mean(x[idx], vth);
        }
    }
}

extern "C" void kernel_launch(void* const* d_in, const int* in_sizes, int n_in,
                              void* d_out, int out_size, void* d_ws, size_t ws_size,
                              hipStream_t stream) {
    const float* x = (const float*)d_in[0];     // [16,128,56,56] fp32
    const float* qthr = (const float*)d_in[1];  // scalar learnable threshold
    float* out = (float*)d_out;                 // same shape as x

    const int n = in_sizes[0];
    const int numTiles = n / TILE_ELEMS;
    int blocks = (numTiles + TPB - 1) / TPB;
    if (blocks < 1) blocks = 1;

    stbif_ms_kernel<<<blocks, BLOCK, 2 * BLOCK * sizeof(v4f), stream>>>(
        x, qthr, out, n, numTiles);
}